// Gating_ROIHeads_4767413698911
// MI455X (gfx1250) — compile-verified
//
#include <hip/hip_runtime.h>
#include <hip/hip_bf16.h>

// ---------------- problem constants (from reference) ----------------
#define IMG_H 800
#define IMG_W 1344
#define FH 50
#define FW 84
#define HW 4200            // FH*FW
#define CCH 1024
#define N_PROP 3000
#define NTOT 6000
#define K_KEEP 512
#define POOL 14
#define PP 196             // POOL*POOL
#define NCLS1 81           // NUM_CLASSES+1
#define NCLS1_PAD 96       // padded to multiple of 16
#define NBOX 320           // NUM_CLASSES*4 (already multiple of 16)
#define LVL_OFF 1345.0f    // float(max(IMG_H,IMG_W)+1)
#define NEG_INF (-__builtin_inff())

// ---------------- output layout (floats) ----------------
#define OFF_BF1  ((size_t)0)
#define OFF_BF2  ((size_t)102760448)                 // 512*1024*196
#define OFF_S1   ((size_t)205520896)
#define OFF_S2   (OFF_S1 + (size_t)(K_KEEP*NCLS1))   // +41472
#define OFF_D1   (OFF_S2 + (size_t)(K_KEEP*NCLS1))
#define OFF_PROP (OFF_D1 + (size_t)(K_KEEP*NBOX))
#define OFF_FEAT (OFF_PROP + (size_t)(K_KEEP*4))

// ---------------- workspace layout (floats) ----------------
#define WS_BOXOFF  ((size_t)0)                       // 6000*4
#define WS_BOXCLIP ((size_t)24000)                   // 6000*4
#define WS_SCORES  ((size_t)48000)                   // 6000
#define WS_X1      ((size_t)54528)                   // 512*1024  (8B aligned)
#define WS_X2      (WS_X1 + (size_t)(K_KEEP*CCH))
#define WS_PROP    (WS_X2 + (size_t)(K_KEEP*CCH))    // 512*4
#define WS_BT_C1   (WS_PROP + (size_t)2048)          // 1024*96 packed W_cls1
#define WS_BT_C2   (WS_BT_C1 + (size_t)(CCH*NCLS1_PAD))
#define WS_BT_B1   (WS_BT_C2 + (size_t)(CCH*NCLS1_PAD))  // 1024*320 packed W_box1

typedef __attribute__((ext_vector_type(2))) float v2f;
typedef __attribute__((ext_vector_type(8))) float v8f;

// ============ 1) clip boxes, add level offsets, concat scores ============
__global__ __launch_bounds__(256)
void prep_kernel(const float* __restrict__ b1, const float* __restrict__ b2,
                 const float* __restrict__ l1, const float* __restrict__ l2,
                 const int* __restrict__ lev1, const int* __restrict__ lev2,
                 float* __restrict__ box_off, float* __restrict__ box_clip,
                 float* __restrict__ scores) {
  int i = blockIdx.x * blockDim.x + threadIdx.x;
  if (i >= NTOT) return;
  const float* src = (i < N_PROP) ? (b1 + (size_t)i * 4) : (b2 + (size_t)(i - N_PROP) * 4);
  float x0 = fminf(fmaxf(src[0], 0.f), (float)IMG_W);
  float y0 = fminf(fmaxf(src[1], 0.f), (float)IMG_H);
  float x1 = fminf(fmaxf(src[2], 0.f), (float)IMG_W);
  float y1 = fminf(fmaxf(src[3], 0.f), (float)IMG_H);
  int lv   = (i < N_PROP) ? lev1[i] : lev2[i - N_PROP];
  float off = (float)lv * LVL_OFF;
  box_clip[i * 4 + 0] = x0;  box_clip[i * 4 + 1] = y0;
  box_clip[i * 4 + 2] = x1;  box_clip[i * 4 + 3] = y1;
  box_off[i * 4 + 0] = x0 + off;  box_off[i * 4 + 1] = y0 + off;
  box_off[i * 4 + 2] = x1 + off;  box_off[i * 4 + 3] = y1 + off;
  scores[i] = (i < N_PROP) ? l1[i] : l2[i - N_PROP];
}

// ============ 1b) pack weights: B[1024,N] -> Bt[(k/2)*Npad + col]{k,k+1}, zero-padded ====
// Gives each WMMA lane a single contiguous 8B load per K-step (global_load_b64).
__global__ __launch_bounds__(256)
void pack_b_kernel(const float* __restrict__ B, float* __restrict__ Bt, int N, int Npad) {
  int idx = blockIdx.x * blockDim.x + threadIdx.x;   // over (K/2)*Npad
  int total = (CCH / 2) * Npad;
  if (idx >= total) return;
  int p   = idx / Npad;        // K-pair index
  int col = idx - p * Npad;
  int k = p * 2;
  float b0 = (col < N) ? B[(size_t)k * N + col] : 0.f;
  float b1 = (col < N) ? B[(size_t)(k + 1) * N + col] : 0.f;
  Bt[(size_t)idx * 2 + 0] = b0;
  Bt[(size_t)idx * 2 + 1] = b1;
}

// ============ 2) greedy NMS: single 32-wave block, scores in LDS ============
__global__ __launch_bounds__(1024)
void nms_kernel(const float* __restrict__ box_off, const float* __restrict__ box_clip,
                const float* __restrict__ scores_in,
                float* __restrict__ prop_out, float* __restrict__ prop_ws) {
  __shared__ float ss[NTOT];
  __shared__ float rv[32];
  __shared__ int   ri[32];
  __shared__ int   sbest;
  const int t = threadIdx.x;
  const int lane = t & 31, wv = t >> 5;
  for (int i = t; i < NTOT; i += 1024) ss[i] = scores_in[i];
  __syncthreads();

  for (int it = 0; it < K_KEEP; ++it) {
    // ---- argmax over LDS scores (lowest index on ties, like jnp.argmax) ----
    float bvv = NEG_INF; int bii = NTOT;
    for (int i = t; i < NTOT; i += 1024) {
      float v = ss[i];
      if (v > bvv || (v == bvv && i < bii)) { bvv = v; bii = i; }
    }
    #pragma unroll
    for (int d = 16; d > 0; d >>= 1) {
      float ov = __shfl_down(bvv, d);
      int   oi = __shfl_down(bii, d);
      if (ov > bvv || (ov == bvv && oi < bii)) { bvv = ov; bii = oi; }
    }
    if (lane == 0) { rv[wv] = bvv; ri[wv] = bii; }
    __syncthreads();
    if (wv == 0) {
      bvv = rv[lane]; bii = ri[lane];
      #pragma unroll
      for (int d = 16; d > 0; d >>= 1) {
        float ov = __shfl_down(bvv, d);
        int   oi = __shfl_down(bii, d);
        if (ov > bvv || (ov == bvv && oi < bii)) { bvv = ov; bii = oi; }
      }
      if (lane == 0) {
        sbest = bii;
        ss[bii] = NEG_INF;
        float p0 = box_clip[bii * 4 + 0], p1 = box_clip[bii * 4 + 1];
        float p2 = box_clip[bii * 4 + 2], p3 = box_clip[bii * 4 + 3];
        prop_out[it * 4 + 0] = p0; prop_out[it * 4 + 1] = p1;
        prop_out[it * 4 + 2] = p2; prop_out[it * 4 + 3] = p3;
        prop_ws [it * 4 + 0] = p0; prop_ws [it * 4 + 1] = p1;
        prop_ws [it * 4 + 2] = p2; prop_ws [it * 4 + 3] = p3;
      }
    }
    __syncthreads();
    // ---- suppress ----
    const int best = sbest;
    const float bx0 = box_off[best * 4 + 0], by0 = box_off[best * 4 + 1];
    const float bx1 = box_off[best * 4 + 2], by1 = box_off[best * 4 + 3];
    const float barea = (bx1 - bx0) * (by1 - by0);
    for (int i = t; i < NTOT; i += 1024) {
      if (ss[i] == NEG_INF) continue;
      float x0 = box_off[i * 4 + 0], y0 = box_off[i * 4 + 1];
      float x1 = box_off[i * 4 + 2], y1 = box_off[i * 4 + 3];
      float ix0 = fmaxf(bx0, x0), iy0 = fmaxf(by0, y0);
      float ix1 = fminf(bx1, x1), iy1 = fminf(by1, y1);
      float inter = fmaxf(ix1 - ix0, 0.f) * fmaxf(iy1 - iy0, 0.f);
      float area  = (x1 - x0) * (y1 - y0);
      float iou = inter / (barea + area - inter + 1e-9f);
      if (iou > 0.7f) ss[i] = NEG_INF;
    }
    __syncthreads();
  }
}

// ====== 3) fused ROIAlign + both affine heads + channel means (1 block/ROI) ======
__global__ __launch_bounds__(256)
void roi_head_kernel(const float* __restrict__ feat, const float* __restrict__ prop,
                     const float* __restrict__ scale1, const float* __restrict__ bias1,
                     const float* __restrict__ scale2, const float* __restrict__ bias2,
                     float* __restrict__ bf1, float* __restrict__ bf2,
                     float* __restrict__ x1o, float* __restrict__ x2o) {
  const int r = blockIdx.x;
  const int t = threadIdx.x;
  __shared__ int   sxa[POOL], sxb[POOL], sya[POOL], syb[POOL];
  __shared__ float slx[POOL], sly[POOL];
  __shared__ int   i00[PP], i01[PP], i10[PP], i11[PP];
  __shared__ float w00[PP], w01[PP], w10[PP], w11[PP];

  const float px0 = prop[r * 4 + 0] * (1.f / 16.f) - 0.5f;
  const float py0 = prop[r * 4 + 1] * (1.f / 16.f) - 0.5f;
  const float px1 = prop[r * 4 + 2] * (1.f / 16.f) - 0.5f;
  const float py1 = prop[r * 4 + 3] * (1.f / 16.f) - 0.5f;
  const float bw = (px1 - px0) * (1.f / POOL);
  const float bh = (py1 - py0) * (1.f / POOL);

  if (t < POOL) {
    float xs = px0 + ((float)t + 0.5f) * bw;
    float xf = floorf(xs);
    slx[t] = xs - xf;
    int xi = (int)xf;
    sxa[t] = min(max(xi, 0), FW - 1);
    sxb[t] = min(max(xi + 1, 0), FW - 1);
  } else if (t < 2 * POOL) {
    int j = t - POOL;
    float ys = py0 + ((float)j + 0.5f) * bh;
    float yf = floorf(ys);
    sly[j] = ys - yf;
    int yi = (int)yf;
    sya[j] = min(max(yi, 0), FH - 1);
    syb[j] = min(max(yi + 1, 0), FH - 1);
  }
  __syncthreads();
  if (t < PP) {
    int py = t / POOL, px = t % POOL;
    float ly = sly[py], lx = slx[px];
    float wy0 = 1.f - ly, wy1 = ly, wx0 = 1.f - lx, wx1 = lx;
    i00[t] = sya[py] * FW + sxa[px];  w00[t] = wy0 * wx0;
    i01[t] = sya[py] * FW + sxb[px];  w01[t] = wy0 * wx1;
    i10[t] = syb[py] * FW + sxa[px];  w10[t] = wy1 * wx0;
    i11[t] = syb[py] * FW + sxb[px];  w11[t] = wy1 * wx1;
  }
  __syncthreads();

  const int wave = t >> 5, lane = t & 31;
  for (int c = wave; c < CCH; c += 8) {
    const float* fc = feat + (size_t)c * HW;
    const float s1c = scale1[c], b1c = bias1[c];
    const float s2c = scale2[c], b2c = bias2[c];
    float sum = 0.f;
    const size_t base = ((size_t)r * CCH + (size_t)c) * PP;
    for (int p = lane; p < PP; p += 32) {
      float v = fc[i00[p]] * w00[p] + fc[i01[p]] * w01[p]
              + fc[i10[p]] * w10[p] + fc[i11[p]] * w11[p];
      sum += v;
      bf1[base + p] = fmaf(v, s1c, b1c);
      bf2[base + p] = fmaf(v, s2c, b2c);
    }
    #pragma unroll
    for (int d = 16; d > 0; d >>= 1) sum += __shfl_down(sum, d);
    if (lane == 0) {
      float m = sum * (1.f / (float)PP);
      x1o[(size_t)r * CCH + c] = fmaf(m, s1c, b1c);
      x2o[(size_t)r * CCH + c] = fmaf(m, s2c, b2c);
    }
  }
}

// ============ 4) f32 WMMA GEMM on packed B: out[512,N] = A @ B + bias ============
// One wave per 16x16 output tile; V_WMMA_F32_16X16X4_F32, 256 K-steps.
// Branch-free inner loop: one global_load_b64 for A frag, one for B frag, one WMMA.
// Lane mapping per CDNA5 ISA 7.12.2:
//   A (16x4 f32): lanes 0-15 M=lane with K{0,1}; lanes 16-31 K{2,3}.
//   B (4x16 f32): lanes 0-15 N=lane with K{0,1}; lanes 16-31 K{2,3}  (packed pairs).
//   C/D 16x16 f32: VGPR j = row (j or 8+j), N = lane&15.
__global__ __launch_bounds__(256)
void gemm_wmma_kernel(const float* __restrict__ A, const float* __restrict__ Bt,
                      const float* __restrict__ bias, float* __restrict__ out,
                      int N, int Npad, int ntiles) {
  const int wave = threadIdx.x >> 5, lane = threadIdx.x & 31;
  const int tm = blockIdx.x;
  const int tn = blockIdx.y * 8 + wave;
  if (tn >= ntiles) return;
  const int hf  = lane >> 4;        // lane half: selects K pair
  const int l15 = lane & 15;
  const int rowA = tm * 16 + l15;
  const int col  = tn * 16 + l15;   // always < Npad (zero-padded)
  const float* pA = A + (size_t)rowA * CCH + (size_t)(hf * 2);
  const size_t strideB = (size_t)Npad * 2;          // floats per K-pair row
  const float* pB = Bt + ((size_t)hf * Npad + col) * 2;

  v8f acc = {0.f, 0.f, 0.f, 0.f, 0.f, 0.f, 0.f, 0.f};
  #pragma unroll 8
  for (int k = 0; k < CCH; k += 4) {
    v2f a = *(const v2f*)(pA + k);
    v2f b = *(const v2f*)(pB + (size_t)(k >> 1) * strideB);
    __builtin_prefetch(pB + (size_t)((k >> 1) + 16) * strideB, 0, 1);  // speculative
    acc = __builtin_amdgcn_wmma_f32_16x16x4_f32(
        /*neg_a=*/false, a, /*neg_b=*/false, b,
        /*c_mod=*/(short)0, acc, /*reuse_a=*/false, /*reuse_b=*/false);
  }
  if (col < N) {
    const float bv = bias[col];
    const int mrow = tm * 16 + hf * 8;
    size_t o = (size_t)mrow * N + col;
    #pragma unroll
    for (int j = 0; j < 8; ++j) out[o + (size_t)j * N] = acc[j] + bv;
  }
}

// =========================== launch ===========================
extern "C" void kernel_launch(void* const* d_in, const int* in_sizes, int n_in,
                              void* d_out, int out_size, void* d_ws, size_t ws_size,
                              hipStream_t stream) {
  const float* feat   = (const float*)d_in[0];
  const float* boxes1 = (const float*)d_in[1];
  const float* boxes2 = (const float*)d_in[2];
  const float* logit1 = (const float*)d_in[3];
  const float* logit2 = (const float*)d_in[4];
  const int*   lev1   = (const int*)d_in[5];
  const int*   lev2   = (const int*)d_in[6];
  const float* scale1 = (const float*)d_in[7];
  const float* bias1  = (const float*)d_in[8];
  const float* scale2 = (const float*)d_in[9];
  const float* bias2  = (const float*)d_in[10];
  const float* Wc1    = (const float*)d_in[11];
  const float* bc1    = (const float*)d_in[12];
  const float* Wb1    = (const float*)d_in[13];
  const float* bb1    = (const float*)d_in[14];
  const float* Wc2    = (const float*)d_in[15];
  const float* bc2    = (const float*)d_in[16];

  float* out = (float*)d_out;
  float* ws  = (float*)d_ws;

  prep_kernel<<<(NTOT + 255) / 256, 256, 0, stream>>>(
      boxes1, boxes2, logit1, logit2, lev1, lev2,
      ws + WS_BOXOFF, ws + WS_BOXCLIP, ws + WS_SCORES);

  // pack weight panels (independent of NMS; overlaps nicely in-order)
  pack_b_kernel<<<((CCH / 2) * NCLS1_PAD + 255) / 256, 256, 0, stream>>>(
      Wc1, ws + WS_BT_C1, NCLS1, NCLS1_PAD);
  pack_b_kernel<<<((CCH / 2) * NCLS1_PAD + 255) / 256, 256, 0, stream>>>(
      Wc2, ws + WS_BT_C2, NCLS1, NCLS1_PAD);
  pack_b_kernel<<<((CCH / 2) * NBOX + 255) / 256, 256, 0, stream>>>(
      Wb1, ws + WS_BT_B1, NBOX, NBOX);

  nms_kernel<<<1, 1024, 0, stream>>>(
      ws + WS_BOXOFF, ws + WS_BOXCLIP, ws + WS_SCORES,
      out + OFF_PROP, ws + WS_PROP);

  roi_head_kernel<<<K_KEEP, 256, 0, stream>>>(
      feat, ws + WS_PROP, scale1, bias1, scale2, bias2,
      out + OFF_BF1, out + OFF_BF2, ws + WS_X1, ws + WS_X2);

  // s1 = x1 @ W_cls1 + b_cls1   (Npad=96 -> 6 tiles)
  gemm_wmma_kernel<<<dim3(32, 1), 256, 0, stream>>>(
      ws + WS_X1, ws + WS_BT_C1, bc1, out + OFF_S1, NCLS1, NCLS1_PAD, 6);
  // s2 = x2 @ W_cls2 + b_cls2
  gemm_wmma_kernel<<<dim3(32, 1), 256, 0, stream>>>(
      ws + WS_X2, ws + WS_BT_C2, bc2, out + OFF_S2, NCLS1, NCLS1_PAD, 6);
  // d1 = x1 @ W_box1 + b_box1   (N=320 -> 20 tiles -> grid.y=3)
  gemm_wmma_kernel<<<dim3(32, 3), 256, 0, stream>>>(
      ws + WS_X1, ws + WS_BT_B1, bb1, out + OFF_D1, NBOX, NBOX, 20);

  // feat passthrough output
  hipMemcpyAsync(out + OFF_FEAT, feat, (size_t)CCH * HW * sizeof(float),
                 hipMemcpyDeviceToDevice, stream);
}